// Safe_Net_model_71923522339410
// MI455X (gfx1250) — compile-verified
//
#include <hip/hip_runtime.h>
#include <hip/hip_bf16.h>
#include <math.h>

typedef __attribute__((ext_vector_type(2))) float v2f;
typedef __attribute__((ext_vector_type(8))) float v8f;

#define BSZ       128
#define CC        768
#define BOT       512
#define FEAT_ROW  (257 * 768)   // row stride of features per batch
#define NPIX      256

// ---------------------------------------------------------------------------
// Kernel 1: localization MLP + affine grid -> nearest-neighbor gather indices
// ---------------------------------------------------------------------------
__global__ void loc_grid_kernel(const float* __restrict__ features,
                                const float* __restrict__ locW1,
                                const float* __restrict__ locb1,
                                const float* __restrict__ locW2,
                                const float* __restrict__ locb2,
                                int* __restrict__ lin) {
  const int b = blockIdx.x;
  const int t = threadIdx.x;  // 128 threads
  __shared__ float gf[CC];
  __shared__ float hbuf[128];
  __shared__ float theta[6];

  const float* grow = features + (size_t)b * FEAT_ROW;  // token 0
  for (int i = t; i < CC; i += 128) gf[i] = grow[i];
  __syncthreads();

  {
    const float* w = locW1 + (size_t)t * CC;
    float acc = locb1[t];
#pragma unroll 4
    for (int k = 0; k < CC; ++k) acc = fmaf(gf[k], w[k], acc);
    hbuf[t] = fmaxf(acc, 0.f);
  }
  __syncthreads();

  if (t < 6) {
    const float* w = locW2 + t * 128;
    float acc = locb2[t];
#pragma unroll 4
    for (int k = 0; k < 128; ++k) acc = fmaf(hbuf[k], w[k], acc);
    theta[t] = acc;
  }
  __syncthreads();

  for (int p = t; p < NPIX; p += 128) {
    const int hi = p >> 4, wi = p & 15;
    const float Xc = -1.f + (2.f / 15.f) * (float)wi;
    const float Yc = -1.f + (2.f / 15.f) * (float)hi;
    const float gx = theta[0] * Xc + theta[1] * Yc + theta[2];
    const float gy = theta[3] * Xc + theta[4] * Yc + theta[5];
    const float fx = rintf((gx + 1.f) * 0.5f * 15.f);  // round half-to-even == jnp.round
    const float fy = rintf((gy + 1.f) * 0.5f * 15.f);
    const int ix = (int)fminf(fmaxf(fx, 0.f), 15.f);
    const int iy = (int)fminf(fmaxf(fy, 0.f), 15.f);
    lin[b * NPIX + p] = iy * 16 + ix;
  }
}

// ---------------------------------------------------------------------------
// Kernel 2: gather + channel-max + ring sums + boundary pick + partition sums
// One block per batch, 256 threads, 3 channels per thread. Single pass over
// the 768KB gathered tile.
// ---------------------------------------------------------------------------
__device__ __forceinline__ int pick_boundary(const float* sa, int center, int margin, int n) {
  const float cv = sa[center], mv = sa[margin];
  float best = INFINITY;
  int bi = 0;
#pragma unroll
  for (int i = 0; i < 8; ++i) {
    float f = fabsf(fabsf(sa[i] - cv) - fabsf(sa[i] - mv));
    if (i < center || i > margin) f = INFINITY;
    if (f < best) { best = f; bi = i; }  // first min, matches jnp.argmin
  }
  const int lo = center + (1 << (n - 1));
  const int hi = margin - (1 << (n - 1)) + 1;
  int v = bi < lo ? lo : bi;            // clip = min(max(b,lo),hi)
  v = v > hi ? hi : v;
  return v;
}

__device__ __forceinline__ int ring_of(int p) {  // 1..8
  const int hh = p >> 4, ww = p & 15;
  const int dh = hh >= 8 ? hh - 7 : 8 - hh;
  const int dw = ww >= 8 ? ww - 7 : 8 - ww;
  return dh > dw ? dh : dw;
}

__global__ void gather_rings_kernel(const float* __restrict__ features,
                                    const int* __restrict__ lin,
                                    float* __restrict__ partBuf) {
  const int b = blockIdx.x;
  const int t = threadIdx.x;  // 0..255
  __shared__ int      slin[NPIX];
  __shared__ int      order[NPIX];    // pixel ids grouped by ring; start of ring r = (2r)^2
  __shared__ unsigned aggKey[NPIX];   // monotonic-uint encoded per-pixel channel max
  __shared__ float    ringAgg[8];
  __shared__ int      cnt[8];
  __shared__ int      bnds[4];

  slin[t]   = lin[b * NPIX + t];
  aggKey[t] = 0x007FFFFFu;  // encoded -inf
  if (t < 8) { ringAgg[t] = 0.f; cnt[t] = 0; }
  __syncthreads();

  {
    const int r = ring_of(t);
    const int base = (2 * (r - 1)) * (2 * (r - 1));
    const int slot = base + atomicAdd(&cnt[r - 1], 1);
    order[slot] = t;
  }
  __syncthreads();

  const float* __restrict__ pbase = features + (size_t)b * FEAT_ROW + CC;  // patch tokens
  float s0[8], s1[8], s2[8];
#pragma unroll
  for (int r = 0; r < 8; ++r) {
    float a0 = 0.f, a1 = 0.f, a2 = 0.f;
    const int q0 = (2 * r) * (2 * r);
    const int q1 = (2 * r + 2) * (2 * r + 2);
    for (int q = q0; q < q1; ++q) {
      const int p = order[q];
      const float* row = pbase + (size_t)slin[p] * CC;
      const float v0 = row[t];
      const float v1 = row[t + 256];
      const float v2 = row[t + 512];
      a0 += v0; a1 += v1; a2 += v2;
      const float m = fmaxf(fmaxf(v0, v1), v2);
      const unsigned u = __float_as_uint(m);
      const unsigned key = (u & 0x80000000u) ? ~u : (u | 0x80000000u);
      atomicMax(&aggKey[p], key);
    }
    s0[r] = a0; s1[r] = a1; s2[r] = a2;
  }
  __syncthreads();

  {  // per-pixel channel max -> per-ring saliency sums
    const unsigned key = aggKey[t];
    const unsigned u = (key & 0x80000000u) ? (key & 0x7FFFFFFFu) : ~key;
    const float agg = __uint_as_float(u);
    atomicAdd(&ringAgg[ring_of(t) - 1], agg);
  }
  __syncthreads();

  if (t == 0) {
    float sal[8];
#pragma unroll
    for (int i = 0; i < 8; ++i) {
      const float d = 2.f * (float)(i + 1);
      sal[i] = ringAgg[i] / (d * d);
    }
    const int b1 = pick_boundary(sal, 0, 7, 2);
    const int b0 = pick_boundary(sal, 0, b1 - 1, 1);
    const int b2 = pick_boundary(sal, b1, 7, 1);
    bnds[0] = b0; bnds[1] = b1; bnds[2] = b2; bnds[3] = 8;
  }
  __syncthreads();

  // prefix crop-sums P(v) = sum of rings 1..v, then partition outputs
  float p0[9], p1[9], p2[9];
  p0[0] = p1[0] = p2[0] = 0.f;
#pragma unroll
  for (int r = 0; r < 8; ++r) {
    p0[r + 1] = p0[r] + s0[r];
    p1[r + 1] = p1[r] + s1[r];
    p2[r + 1] = p2[r] + s2[r];
  }
#pragma unroll
  for (int i = 0; i < 4; ++i) {
    const int bv = bnds[i];
    const int pv = (i == 0) ? 0 : bnds[i - 1];
    const float den = (2.f * (float)bv) * (2.f * (float)bv);
    float* dst = partBuf + (size_t)i * BSZ * CC + (size_t)b * CC;
    dst[t]       = (p0[bv] - p0[pv]) / den;
    dst[t + 256] = (p1[bv] - p1[pv]) / den;
    dst[t + 512] = (p2[bv] - p2[pv]) / den;
  }
}

// ---------------------------------------------------------------------------
// Kernel 3: five class blocks as WMMA f32 GEMMs + fused bias/BN epilogue.
// 1280 tiles (5 slices x 8 Mtiles x 32 Ntiles), one 16x16 tile per wave.
// A: 16x4 fp32 (lane 0-15 = M row; VGPR pair = K 0/1 for lanes<16, 2/3 else)
// B: 4x16 fp32 (same half-wave K split, lane%16 = N column)
// D: VGPR j holds M = j (lanes 0-15) / j+8 (lanes 16-31), N = lane%16
// ---------------------------------------------------------------------------
__global__ void cls_wmma_kernel(const float* __restrict__ features,
                                const float* __restrict__ partBuf,
                                const float* __restrict__ clsW,
                                const float* __restrict__ clsb,
                                const float* __restrict__ bn_g,
                                const float* __restrict__ bn_b,
                                const float* __restrict__ bn_rm,
                                const float* __restrict__ bn_rv,
                                float* __restrict__ out) {
  const int lane  = threadIdx.x & 31;
  const int wave  = threadIdx.x >> 5;
  const int tile  = blockIdx.x * 8 + wave;  // 0..1279
  const int slice = tile >> 8;              // 0..4
  const int rem   = tile & 255;
  const int m0    = (rem >> 5) << 4;        // batch tile origin
  const int n0    = (rem & 31) << 4;        // neuron tile origin

  const float* X;
  size_t ldx;
  if (slice < 4) { X = partBuf + (size_t)slice * BSZ * CC; ldx = CC; }
  else           { X = features;                            ldx = FEAT_ROW; }

  const int lrow  = lane & 15;
  const int khalf = (lane >> 4) << 1;  // 0 or 2
  const float* xptr = X + (size_t)(m0 + lrow) * ldx + khalf;
  const float* wptr = clsW + ((size_t)slice * BOT + (size_t)(n0 + lrow)) * CC + khalf;

  v8f acc = {};
  for (int k = 0; k < CC; k += 4) {
    const float2 av = *(const float2*)(xptr + k);
    const float2 bv = *(const float2*)(wptr + k);
    v2f a;  a.x = av.x;  a.y = av.y;
    v2f bb; bb.x = bv.x; bb.y = bv.y;
    acc = __builtin_amdgcn_wmma_f32_16x16x4_f32(
        /*neg_a=*/false, a, /*neg_b=*/false, bb,
        /*c_mod=*/(short)0, acc, /*reuse_a=*/false, /*reuse_b=*/false);
  }

  const int n = n0 + lrow;
  const size_t sn = (size_t)slice * BOT + (size_t)n;
  const float bias  = clsb[sn];
  const float scale = bn_g[sn] / sqrtf(bn_rv[sn] + 1e-5f);
  const float rm    = bn_rm[sn];
  const float be    = bn_b[sn];
  const int mhi = (lane >> 4) << 3;  // 0 or 8

#pragma unroll
  for (int j = 0; j < 8; ++j) {
    const int m = m0 + j + mhi;
    const float h = acc[j] + bias;
    out[((size_t)m * BOT + (size_t)n) * 5 + (size_t)slice] = (h - rm) * scale + be;
  }
}

// ---------------------------------------------------------------------------
extern "C" void kernel_launch(void* const* d_in, const int* in_sizes, int n_in,
                              void* d_out, int out_size, void* d_ws, size_t ws_size,
                              hipStream_t stream) {
  const float* features = (const float*)d_in[0];
  const float* locW1    = (const float*)d_in[1];
  const float* locb1    = (const float*)d_in[2];
  const float* locW2    = (const float*)d_in[3];
  const float* locb2    = (const float*)d_in[4];
  const float* clsW     = (const float*)d_in[5];
  const float* clsb     = (const float*)d_in[6];
  const float* bn_g     = (const float*)d_in[7];
  const float* bn_b     = (const float*)d_in[8];
  const float* bn_rm    = (const float*)d_in[9];
  const float* bn_rv    = (const float*)d_in[10];
  float* out = (float*)d_out;

  int*   lin     = (int*)d_ws;                                        // 128*256 ints
  float* partBuf = (float*)((char*)d_ws + BSZ * NPIX * sizeof(int));  // 4*128*768 floats

  loc_grid_kernel<<<BSZ, 128, 0, stream>>>(features, locW1, locb1, locW2, locb2, lin);
  gather_rings_kernel<<<BSZ, 256, 0, stream>>>(features, lin, partBuf);
  cls_wmma_kernel<<<160, 256, 0, stream>>>(features, partBuf, clsW, clsb,
                                           bn_g, bn_b, bn_rm, bn_rv, out);
}